// GCN_39204461478219
// MI455X (gfx1250) — compile-verified
//
#include <hip/hip_runtime.h>

typedef __attribute__((ext_vector_type(2))) float v2f;
typedef __attribute__((ext_vector_type(8))) float v8f;

#define HID 128
#define NGRAPH 64
#define OUTF 32
#define POOL_CHUNK 512

// ---------------- degree / normalization ----------------
__global__ void k_fill1(float* __restrict__ p, int n) {
  int i = blockIdx.x * blockDim.x + threadIdx.x;
  if (i < n) p[i] = 1.0f;  // self-loop contributes 1 to every degree
}

__global__ void k_degcount(const long long* __restrict__ dst, float* __restrict__ deg, int E) {
  int e = blockIdx.x * blockDim.x + threadIdx.x;
  if (e < E) atomicAdd(&deg[dst[e]], 1.0f);
}

__global__ void k_rsqrt(float* __restrict__ d, int n) {
  int i = blockIdx.x * blockDim.x + threadIdx.x;
  if (i < n) d[i] = rsqrtf(d[i]);
}

__global__ void k_zero(float* __restrict__ p, int n) {
  int i = blockIdx.x * blockDim.x + threadIdx.x;
  if (i < n) p[i] = 0.0f;
}

// ---------------- fp32 WMMA GEMM: C[nrows x 128] = act(A[nrows x 128]) @ W[128 x 128] ----------------
// 256 threads = 8 waves; block handles 128 rows (16 per wave); W staged in LDS.
__global__ __launch_bounds__(256)
void k_gemm128(const float* __restrict__ A, const float* __restrict__ W,
               float* __restrict__ C, int nrows, int relu_in) {
  __shared__ float sW[HID * HID];  // 64 KB of the 320 KB WGP LDS
  const int tid = threadIdx.x;

  // cooperative 128x128 fp32 weight stage: 16 float4 per thread
  {
    const float4* __restrict__ Wv = (const float4*)W;
    float4* Sv = (float4*)sW;
#pragma unroll
    for (int i = tid; i < HID * HID / 4; i += 256) Sv[i] = Wv[i];
  }
  __syncthreads();

  const int wave = tid >> 5;
  const int lane = tid & 31;
  const int m = lane & 15;            // M (A) / N (B,C) index within tile
  const int kh = (lane >> 4) << 1;    // K sub-offset: lanes 0-15 -> K{0,1}, 16-31 -> K{2,3}
  const int row0 = blockIdx.x * 128 + wave * 16;

  const int ra = row0 + m;
  const bool in_a = ra < nrows;
  const float* __restrict__ arow = A + (size_t)(in_a ? ra : (nrows - 1)) * HID;

  v8f acc[8] = {};  // 8 column tiles -> full 128-wide output per wave

  for (int k0 = 0; k0 < HID; k0 += 4) {
    float a0 = arow[k0 + kh];
    float a1 = arow[k0 + kh + 1];
    if (!in_a) { a0 = 0.0f; a1 = 0.0f; }
    if (relu_in) { a0 = fmaxf(a0, 0.0f); a1 = fmaxf(a1, 0.0f); }
    v2f av; av.x = a0; av.y = a1;
#pragma unroll
    for (int c = 0; c < 8; ++c) {
      v2f bv;
      bv.x = sW[(k0 + kh) * HID + c * 16 + m];
      bv.y = sW[(k0 + kh + 1) * HID + c * 16 + m];
      acc[c] = __builtin_amdgcn_wmma_f32_16x16x4_f32(
          /*neg_a=*/false, av, /*neg_b=*/false, bv,
          /*c_mod=*/(short)0, acc[c], /*reuse_a=*/false, /*reuse_b=*/false);
    }
  }

  // C/D layout: lanes 0-15 -> M=r, lanes 16-31 -> M=8+r; N = lane&15
  const int rstore = row0 + (lane >> 4) * 8;
#pragma unroll
  for (int c = 0; c < 8; ++c) {
#pragma unroll
    for (int r = 0; r < 8; ++r) {
      const int rr = rstore + r;
      if (rr < nrows) C[(size_t)rr * HID + c * 16 + m] = acc[c][r];
    }
  }
}

// ---------------- aggregation ----------------
// out[i] = h[i] * dinv[i]^2 + bias   (self-loop term; also fully initializes out)
__global__ void k_agg_init(const float* __restrict__ h, const float* __restrict__ dinv,
                           const float* __restrict__ bias, float* __restrict__ out, int n) {
  int i = blockIdx.x * blockDim.x + threadIdx.x;  // over n*128 (12.8M, fits int)
  if (i < n * HID) {
    int node = i >> 7;
    int f = i & (HID - 1);
    float di = dinv[node];
    out[i] = h[i] * di * di + bias[f];
  }
}

// scatter-add over edges: out[dst] += h[src] * dinv[src]*dinv[dst]
// 32 lanes per edge, float4 per lane; h rows live in the 192 MB L2.
__global__ void k_agg_edges(const float* __restrict__ h, const float* __restrict__ dinv,
                            const long long* __restrict__ src, const long long* __restrict__ dst,
                            float* __restrict__ out, int E) {
  int t = blockIdx.x * blockDim.x + threadIdx.x;
  int e = t >> 5;
  if (e >= E) return;
  int g = (t & 31) << 2;  // feature offset 0..124
  long long s = src[e];
  long long d = dst[e];
  float norm = dinv[s] * dinv[d];
  const float4 v = *(const float4*)(h + (size_t)s * HID + g);
  float* o = out + (size_t)d * HID + g;
  atomicAdd(o + 0, v.x * norm);
  atomicAdd(o + 1, v.y * norm);
  atomicAdd(o + 2, v.z * norm);
  atomicAdd(o + 3, v.w * norm);
}

// ---------------- pooling (batch is sorted: run-length accumulate, flush per run) ----------------
__global__ __launch_bounds__(128)
void k_pool(const float* __restrict__ h, const long long* __restrict__ batch,
            float* __restrict__ gsum, float* __restrict__ gcnt, int n) {
  const int f = threadIdx.x;  // feature 0..127
  const int start = blockIdx.x * POOL_CHUNK;
  const int end = (start + POOL_CHUNK < n) ? (start + POOL_CHUNK) : n;
  float acc = 0.0f;
  int cur = -1;
  for (int i = start; i < end; ++i) {
    int b = (int)batch[i];
    if (b != cur) {
      if (cur >= 0) atomicAdd(&gsum[cur * HID + f], acc);
      cur = b;
      acc = 0.0f;
    }
    acc += fmaxf(h[(size_t)i * HID + f], 0.0f);  // fused ReLU of layer 2
  }
  if (cur >= 0) atomicAdd(&gsum[cur * HID + f], acc);

  if (f == 0) {  // node counts per graph, same run-length trick
    float c = 0.0f;
    int cb = -1;
    for (int i = start; i < end; ++i) {
      int b = (int)batch[i];
      if (b != cb) {
        if (cb >= 0) atomicAdd(&gcnt[cb], c);
        cb = b;
        c = 0.0f;
      }
      c += 1.0f;
    }
    if (cb >= 0) atomicAdd(&gcnt[cb], c);
  }
}

// ---------------- final FC: out[64 x 32] = (gsum/cnt) @ Wfc + bfc ----------------
__global__ void k_fc(const float* __restrict__ gsum, const float* __restrict__ gcnt,
                     const float* __restrict__ Wfc, const float* __restrict__ bfc,
                     float* __restrict__ out) {
  int t = blockIdx.x * blockDim.x + threadIdx.x;
  if (t >= NGRAPH * OUTF) return;
  int gi = t / OUTF;
  int j = t % OUTF;
  float inv = 1.0f / fmaxf(gcnt[gi], 1.0f);
  float s = 0.0f;
  for (int k = 0; k < HID; ++k) s += gsum[gi * HID + k] * Wfc[k * OUTF + j];
  out[t] = s * inv + bfc[j];
}

extern "C" void kernel_launch(void* const* d_in, const int* in_sizes, int n_in,
                              void* d_out, int out_size, void* d_ws, size_t ws_size,
                              hipStream_t stream) {
  const float* x       = (const float*)d_in[0];
  const long long* ei  = (const long long*)d_in[1];  // [2, E] int64
  const long long* bat = (const long long*)d_in[2];
  const float* W1      = (const float*)d_in[3];
  const float* b1      = (const float*)d_in[4];
  const float* W2      = (const float*)d_in[5];
  const float* b2      = (const float*)d_in[6];
  const float* Wfc     = (const float*)d_in[7];
  const float* bfc     = (const float*)d_in[8];

  const int N = in_sizes[0] / HID;
  const int E = in_sizes[1] / 2;
  const long long* src = ei;
  const long long* dst = ei + E;

  // workspace layout (floats): dinv[N] | h[N*128] | a[N*128] | gsum[64*128] | gcnt[64]
  float* wsf  = (float*)d_ws;
  float* dinv = wsf;
  float* hbuf = wsf + (((size_t)N + 255) & ~(size_t)255);
  float* abuf = hbuf + (size_t)N * HID;
  float* gsum = abuf + (size_t)N * HID;
  float* gcnt = gsum + NGRAPH * HID;

  const int T = 256;
  // degrees -> dinv = rsqrt(deg)  (deg starts at 1 for self-loop)
  k_fill1<<<(N + T - 1) / T, T, 0, stream>>>(dinv, N);
  k_degcount<<<(E + T - 1) / T, T, 0, stream>>>(dst, dinv, E);
  k_rsqrt<<<(N + T - 1) / T, T, 0, stream>>>(dinv, N);
  k_zero<<<(NGRAPH * HID + NGRAPH + T - 1) / T, T, 0, stream>>>(gsum, NGRAPH * HID + NGRAPH);

  const int gemmBlocks = (N + 127) / 128;
  const int nfThreads  = N * HID;  // 12.8M

  // layer 1
  k_gemm128<<<gemmBlocks, 256, 0, stream>>>(x, W1, hbuf, N, 0);
  k_agg_init<<<(nfThreads + T - 1) / T, T, 0, stream>>>(hbuf, dinv, b1, abuf, N);
  k_agg_edges<<<(E * 32 + T - 1) / T, T, 0, stream>>>(hbuf, dinv, src, dst, abuf, E);

  // layer 2 (ReLU of layer-1 output fused into the A-load)
  k_gemm128<<<gemmBlocks, 256, 0, stream>>>(abuf, W2, hbuf, N, 1);
  k_agg_init<<<(nfThreads + T - 1) / T, T, 0, stream>>>(hbuf, dinv, b2, abuf, N);
  k_agg_edges<<<(E * 32 + T - 1) / T, T, 0, stream>>>(hbuf, dinv, src, dst, abuf, E);

  // mean pool (ReLU fused) + FC
  k_pool<<<(N + POOL_CHUNK - 1) / POOL_CHUNK, 128, 0, stream>>>(abuf, bat, gsum, gcnt, N);
  k_fc<<<(NGRAPH * OUTF + T - 1) / T, T, 0, stream>>>(gsum, gcnt, Wfc, bfc, (float*)d_out);
}